// KSLinear_81243601371653
// MI455X (gfx1250) — compile-verified
//
#include <hip/hip_runtime.h>

typedef float v2f __attribute__((ext_vector_type(2)));
typedef float v8f __attribute__((ext_vector_type(8)));
typedef unsigned int v4u __attribute__((ext_vector_type(4)));
typedef int v4i __attribute__((ext_vector_type(4)));
typedef int v8i __attribute__((ext_vector_type(8)));

static constexpr int TOK = 16;    // tokens per workgroup == WMMA M
static constexpr int DIM = 4096;
static constexpr int NW  = 8;     // waves per workgroup

#define WMMA_F32(A, B, C) \
  __builtin_amdgcn_wmma_f32_16x16x4_f32(false, (A), false, (B), (short)0, (C), false, false)

__launch_bounds__(256, 1)
__global__ void ks_fused(const float* __restrict__ x,
                         const float* __restrict__ w0,
                         const float* __restrict__ w1,
                         const float* __restrict__ bias,
                         float* __restrict__ out)
{
    extern __shared__ __align__(16) float tile[];   // TOK * DIM floats = 256 KB

    const int tid  = threadIdx.x;
    const int lane = tid & 31;
    const int wave = tid >> 5;
    const int n    = lane & 15;   // N column (b) lane index
    const int kh   = lane >> 4;   // K-half selector per WMMA f32 layout

    const long token0 = (long)blockIdx.x * TOK;

    // ---- Stage x tile via Tensor Data Mover: 2D tile 4096 x 16 fp32 -> LDS ----
    if (wave == 0) {
        const unsigned lds_base = (unsigned)(size_t)(void*)tile;     // addr[31:0] = LDS offset
        const unsigned long long ga = (unsigned long long)(size_t)(x + token0 * DIM);
        // D# group 0: count=1 | lds_addr | global_addr[56:0] | type=2
        v4u g0 = { 1u,
                   lds_base,
                   (unsigned)(ga & 0xffffffffull),
                   (unsigned)((ga >> 32) & 0x01ffffffull) | 0x80000000u };
        // D# group 1: data_size=4B; tensor_dim0=4096; tensor_dim1=16;
        //             tile_dim0=4096; tile_dim1=16; tensor_dim0_stride=4096
        v8i g1 = { (int)(2u << 16),        // [17:16] data_size=2 (4 bytes); mask/flags=0
                   (int)(4096u << 16),     // [63:48] tensor_dim0[15:0] (abar addr = 0)
                   (int)(16u << 16),       // [79:64] tensor_dim0[31:16]=0 | [95:80] tensor_dim1[15:0]
                   (int)(4096u << 16),     // [111:96] tensor_dim1[31:16]=0 | [127:112] tile_dim0
                   16,                     // [143:128] tile_dim1=16 | [159:144] tile_dim2=0
                   4096,                   // [191:160] tensor_dim0_stride[31:0]
                   0,                      // stride[47:32]=0 | tensor_dim1_stride[15:0]=0
                   0 };
        v4i gz4 = { 0, 0, 0, 0 };          // groups 2/3: higher dims unused
        v8i gz8 = { 0, 0, 0, 0, 0, 0, 0, 0 };
        __builtin_amdgcn_tensor_load_to_lds(g0, g1, gz4, gz4, gz8, 0);
        __builtin_amdgcn_s_wait_tensorcnt(0);
    }
    __syncthreads();

    // ---- Factor 0: y[m,b,d] = sum_c x[m,c,d] * w0[b*4096 + c*64 + d] ----
    // Each wave owns adjacent column pairs (d0, d0+1): contiguous b64 weight
    // loads and b64 LDS writebacks. In-place per-d column.
    #pragma unroll 1
    for (int it = 0; it < 4; ++it) {
        const int d0 = 2 * wave + 16 * it;
        v8f acc[2][4] = {};
        #pragma unroll
        for (int kk = 0; kk < 16; ++kk) {
            const int c0 = kk * 4 + kh * 2;
            v2f q0 = *(const v2f*)&tile[n * DIM + (c0 + 0) * 64 + d0]; // (c0  , d0/d0+1)
            v2f q1 = *(const v2f*)&tile[n * DIM + (c0 + 1) * 64 + d0]; // (c0+1, d0/d0+1)
            v2f a0, a1;
            a0.x = q0.x; a0.y = q1.x;   // A fragment for d0
            a1.x = q0.y; a1.y = q1.y;   // A fragment for d0+1
            #pragma unroll
            for (int j = 0; j < 4; ++j) {
                const float* wp = w0 + (size_t)(j * 16 + n) * DIM + (size_t)c0 * 64 + d0;
                v2f p0 = *(const v2f*)wp;          // (c0  , d0/d0+1)
                v2f p1 = *(const v2f*)(wp + 64);   // (c0+1, d0/d0+1)
                v2f b0, b1;
                b0.x = p0.x; b0.y = p1.x;          // B fragment for d0
                b1.x = p0.y; b1.y = p1.y;          // B fragment for d0+1
                acc[0][j] = WMMA_F32(a0, b0, acc[0][j]);
                acc[1][j] = WMMA_F32(a1, b1, acc[1][j]);
            }
        }
        #pragma unroll
        for (int j = 0; j < 4; ++j)
            #pragma unroll
            for (int v = 0; v < 8; ++v) {
                v2f r; r.x = acc[0][j][v]; r.y = acc[1][j][v];
                *(v2f*)&tile[(v + 8 * kh) * DIM + (j * 16 + n) * 64 + d0] = r;
            }
    }
    __syncthreads();

    // ---- Factor 1: z[m,a,b] = sum_c y[m,a,c] * w1[a*4096 + b*64 + c] ----
    // B pairs (c0, c0+1) are contiguous -> direct v2f loads, no shuffles.
    // In-place per-a block of 64 columns.
    #pragma unroll 1
    for (int ai = wave; ai < 64; ai += NW) {
        v8f acc[4] = {};
        #pragma unroll
        for (int kk = 0; kk < 16; ++kk) {
            const int c0 = kk * 4 + kh * 2;
            v2f a = *(const v2f*)&tile[n * DIM + ai * 64 + c0];
            #pragma unroll
            for (int j = 0; j < 4; ++j) {
                v2f b = *(const v2f*)(w1 + (size_t)ai * DIM + (size_t)(j * 16 + n) * 64 + c0);
                acc[j] = WMMA_F32(a, b, acc[j]);
            }
        }
        #pragma unroll
        for (int j = 0; j < 4; ++j)
            #pragma unroll
            for (int v = 0; v < 8; ++v)
                tile[(v + 8 * kh) * DIM + ai * 64 + (j * 16 + n)] = acc[j][v];
    }
    __syncthreads();

    // ---- Bias add + store: coalesced 128-bit ----
    {
        const float4* b4 = (const float4*)bias;
        const float4* t4 = (const float4*)tile;
        float4* o4 = (float4*)(out + token0 * DIM);
        for (int i = tid; i < TOK * DIM / 4; i += 256) {
            float4 v  = t4[i];
            float4 bb = b4[i & (DIM / 4 - 1)];
            v.x += bb.x; v.y += bb.y; v.z += bb.z; v.w += bb.w;
            o4[i] = v;
        }
    }
}

extern "C" void kernel_launch(void* const* d_in, const int* in_sizes, int n_in,
                              void* d_out, int out_size, void* d_ws, size_t ws_size,
                              hipStream_t stream) {
    (void)in_sizes; (void)n_in; (void)d_ws; (void)ws_size; (void)out_size;
    const float* x    = (const float*)d_in[0];
    const float* w0   = (const float*)d_in[1];
    const float* w1   = (const float*)d_in[2];
    const float* bias = (const float*)d_in[3];
    float* out = (float*)d_out;

    const size_t shmem = (size_t)TOK * DIM * sizeof(float);   // 256 KB (<= 320 KB/WGP)
    (void)hipFuncSetAttribute((const void*)ks_fused,
                              hipFuncAttributeMaxDynamicSharedMemorySize, (int)shmem);

    const int n_tokens = 16384;
    ks_fused<<<n_tokens / TOK, 256, shmem, stream>>>(x, w0, w1, bias, out);
}